// HeteroConv_14147622273721
// MI455X (gfx1250) — compile-verified
//
#include <hip/hip_runtime.h>
#include <hip/hip_bf16.h>
#include <stdint.h>

// ---------------------------------------------------------------------------
// HeteroConv copy_u + segment_sum (gather + scatter-add), MI455X / gfx1250.
//
// dst[edge_dst[e], :] += src_emb[edge_src[e], :]   (D = 64, f32)
//
// Memory-bound: ~256 MB gather + ~256 MB atomic-add, both L2-resident
// (src/dst are 25.6 MB each << 192 MB L2).  Strategy:
//   * 16 lanes per edge, one float4 (b128) gather per lane
//   * fire-and-forget global_atomic_add_f32 scope:SCOPE_DEV (STOREcnt only)
//   * edge index streams double-buffered into LDS with the CDNA5 async
//     global->LDS engine (ASYNCcnt + s_wait_asynccnt), overlapping index
//     fetch with gather/atomic traffic; (src,dst) interleaved so the inner
//     loop reads both with a single ds_load_b64.
// ---------------------------------------------------------------------------

#define TILE   256   // edges staged per LDS tile
#define BLOCK  256   // threads per block (8 wave32)

__global__ void hc_zero_f32(float4* __restrict__ p, int n4) {
    int i      = blockIdx.x * blockDim.x + threadIdx.x;
    int stride = gridDim.x * blockDim.x;
    float4 z = make_float4(0.f, 0.f, 0.f, 0.f);
    for (int j = i; j < n4; j += stride) p[j] = z;
}

// Stage TILE (src,dst) int32 index pairs into LDS, asynchronously.
// Every thread issues one b32 async copy per stream -> each wave's ASYNCcnt
// rises by exactly 2 per stage call (uniform: tail is handled by clamping).
__device__ __forceinline__ void hc_stage(const int* __restrict__ edge_src,
                                         const int* __restrict__ edge_dst,
                                         int* lds_pairs,   // [TILE*2], (s,d) interleaved
                                         int tile, int n_edges) {
    int gidx = tile * TILE + (int)threadIdx.x;
    int cidx = (gidx < n_edges) ? gidx : (n_edges - 1);   // clamp: keep EXEC full
    unsigned goff = (unsigned)cidx * 4u;                  // byte offset into stream
    // Generic pointers to __shared__ carry the LDS byte offset in their low
    // 32 bits (flat aperture rule).
    unsigned lsrc = (unsigned)(uintptr_t)(lds_pairs + 2 * threadIdx.x);
    unsigned ldst = lsrc + 4u;
    asm volatile("global_load_async_to_lds_b32 %0, %1, %2"
                 :: "v"(lsrc), "v"(goff), "s"((uint64_t)(uintptr_t)edge_src)
                 : "memory");
    asm volatile("global_load_async_to_lds_b32 %0, %1, %2"
                 :: "v"(ldst), "v"(goff), "s"((uint64_t)(uintptr_t)edge_dst)
                 : "memory");
}

// One edge slice: gather float4 from src row, 4 device-scope f32 atomic adds.
__device__ __forceinline__ void hc_edge(const float* __restrict__ src_emb,
                                        uint64_t out_base, int s, int d, int l) {
    const float4 v = *reinterpret_cast<const float4*>(
        src_emb + (size_t)s * 64 + (size_t)(l * 4));
    unsigned doff = ((unsigned)d * 64u + (unsigned)(l * 4)) * 4u;
    asm volatile(
        "global_atomic_add_f32 %0, %1, %5 scope:SCOPE_DEV\n\t"
        "global_atomic_add_f32 %0, %2, %5 offset:4 scope:SCOPE_DEV\n\t"
        "global_atomic_add_f32 %0, %3, %5 offset:8 scope:SCOPE_DEV\n\t"
        "global_atomic_add_f32 %0, %4, %5 offset:12 scope:SCOPE_DEV"
        :: "v"(doff), "v"(v.x), "v"(v.y), "v"(v.z), "v"(v.w), "s"(out_base)
        : "memory");
}

__global__ __launch_bounds__(BLOCK)
void hc_scatter_sum(const float* __restrict__ src_emb,
                    const int*   __restrict__ edge_src,
                    const int*   __restrict__ edge_dst,
                    float*       __restrict__ out,
                    int n_edges) {
    __shared__ __align__(8) int s_idx[2][TILE * 2];   // (src,dst) pairs, 2 buffers

    const int ntiles = (n_edges + TILE - 1) / TILE;
    const int g = threadIdx.x >> 4;   // edge slot within a pass   (0..15)
    const int l = threadIdx.x & 15;   // lane within edge: floats [4l, 4l+4)
    const uint64_t out_base = (uint64_t)(uintptr_t)out;

    int buf  = 0;
    int tile = blockIdx.x;
    if (tile < ntiles)
        hc_stage(edge_src, edge_dst, s_idx[buf], tile, n_edges);

    for (; tile < ntiles; tile += (int)gridDim.x) {
        int next = tile + (int)gridDim.x;
        if (next < ntiles) {
            // Prefetch next tile, then drain current tile's 2 async loads.
            // Async loads complete in order, so asynccnt<=2 == tile ready.
            hc_stage(edge_src, edge_dst, s_idx[buf ^ 1], next, n_edges);
            asm volatile("s_wait_asynccnt 0x2" ::: "memory");
        } else {
            asm volatile("s_wait_asynccnt 0x0" ::: "memory");
        }
        __syncthreads();   // staged indices visible to all 8 waves

        const int base = tile * TILE;
        if (base + TILE <= n_edges) {
            // Fast path: full tile, no per-edge predicate.
#pragma unroll 4
            for (int p = 0; p < TILE / 16; ++p) {
                int el = p * 16 + g;
                int2 sd = *reinterpret_cast<const int2*>(&s_idx[buf][2 * el]);
                hc_edge(src_emb, out_base, sd.x, sd.y, l);
            }
        } else {
            // Tail tile (at most one per block).
            for (int p = 0; p < TILE / 16; ++p) {
                int el = p * 16 + g;
                if (base + el < n_edges) {
                    int2 sd = *reinterpret_cast<const int2*>(&s_idx[buf][2 * el]);
                    hc_edge(src_emb, out_base, sd.x, sd.y, l);
                }
            }
        }
        __syncthreads();   // all waves done reading `buf` before restage
        buf ^= 1;
    }
}

extern "C" void kernel_launch(void* const* d_in, const int* in_sizes, int n_in,
                              void* d_out, int out_size, void* d_ws, size_t ws_size,
                              hipStream_t stream) {
    const float* src_emb  = (const float*)d_in[0];
    const int*   edge_src = (const int*)d_in[1];
    const int*   edge_dst = (const int*)d_in[2];
    // d_in[3] = num_dst (device scalar) -- not readable during graph capture;
    // derive sizes from in_sizes/out_size instead.
    float* out = (float*)d_out;

    const int n_edges = in_sizes[1];
    const int n4      = out_size / 4;   // out_size = num_dst * 64, divisible by 4

    hc_zero_f32<<<512, BLOCK, 0, stream>>>((float4*)out, n4);

    const int ntiles = (n_edges + TILE - 1) / TILE;
    int blocks = ntiles < 1024 ? ntiles : 1024;
    if (blocks < 1) blocks = 1;
    hc_scatter_sum<<<blocks, BLOCK, 0, stream>>>(src_emb, edge_src, edge_dst,
                                                 out, n_edges);
}